// RowConv1D_25881472926097
// MI455X (gfx1250) — compile-verified
//
#include <hip/hip_runtime.h>
#include <stdint.h>

typedef float v2f __attribute__((ext_vector_type(2)));
typedef unsigned int u32x4 __attribute__((ext_vector_type(4)));
typedef int i32x4 __attribute__((ext_vector_type(4)));
typedef int i32x8 __attribute__((ext_vector_type(8)));

// Problem constants (match reference)
constexpr int NB = 16;     // batch
constexpr int NT = 2048;   // time
constexpr int ND = 1024;   // channels
constexpr int NS = 32;     // taps

// Tiling
constexpr int TT      = 16;            // output time steps per tile
constexpr int KT      = 8;             // tiles per (persistent) block
constexpr int ROWS    = TT + NS - 1;   // 47 staged input rows per tile
constexpr int CB      = 256;           // channels per block
constexpr int THREADS = 128;           // 4 waves; 2 channels per thread
constexpr int CHUNKS_PER_ROW = CB * 4 / 16;      // 64 b128 chunks per row
constexpr int TILE_ELEMS     = ROWS * CB;        // floats per LDS buffer

#ifndef __has_builtin
#define __has_builtin(x) 0
#endif
#if __has_builtin(__builtin_amdgcn_tensor_load_to_lds) && \
    __has_builtin(__builtin_amdgcn_s_wait_tensorcnt)
#define USE_TDM 1
#else
#define USE_TDM 0
#endif

#if USE_TDM
// Issue one 2D-tile TDM DMA: rows x CB channels x 4B, row stride ND elems.
__device__ __forceinline__ void tdm_issue(const float* gbase, uint32_t lds_addr,
                                          int rows) {
    const uint64_t gaddr = (uint64_t)(uintptr_t)gbase;

    // D# group 0: count=1 | lds_addr | global_addr[56:0] | type=2
    u32x4 g0;
    g0.x = 1u;                                        // count=1, user desc
    g0.y = lds_addr;                                  // lds_addr (bytes)
    g0.z = (uint32_t)(gaddr & 0xFFFFFFFFu);           // global_addr[31:0]
    g0.w = (uint32_t)((gaddr >> 32) & 0x01FFFFFFu)    // global_addr[56:32]
           | (2u << 30);                              // type=2 ("image")

    // D# group 1: data_size=4B; tensor/tile dims; dim0 stride = ND
    i32x8 g1;
    g1[0] = (int)(2u << 16);                          // data_size=2 (4B)
    g1[1] = (int)((uint32_t)CB << 16);                // tensor_dim0[15:0]=256
    g1[2] = (int)(((uint32_t)rows & 0xFFFFu) << 16);  // tensor_dim1[15:0]=rows
    g1[3] = (int)((uint32_t)CB << 16);                // tile_dim0=256
    g1[4] = (int)(uint32_t)rows;                      // tile_dim1=rows, tile_dim2=0
    g1[5] = (int)(uint32_t)ND;                        // tensor_dim0_stride=1024
    g1[6] = 0;
    g1[7] = 0;

    i32x4 g2 = {0, 0, 0, 0};                          // <=2D tensor
    i32x4 g3 = {0, 0, 0, 0};

#if defined(__clang_major__) && (__clang_major__ >= 23)
    i32x8 g4 = {0, 0, 0, 0, 0, 0, 0, 0};
    __builtin_amdgcn_tensor_load_to_lds(g0, g1, g2, g3, g4, 0);
#else
    __builtin_amdgcn_tensor_load_to_lds(g0, g1, g2, g3, 0);
#endif
}
#endif

__global__ __launch_bounds__(THREADS)
void rowconv_tdm_pipe_kernel(const float* __restrict__ x,
                             const float* __restrict__ W,
                             float* __restrict__ out) {
    __shared__ float tile[2 * TILE_ELEMS];   // double buffer: 2 * 47 KiB

    const int tid = threadIdx.x;
    const int bx  = blockIdx.x;
    constexpr int CHUNKS = NT / (TT * KT);   // 16 time chunks per batch row
    const int b     = bx / CHUNKS;
    const int tbase = (bx % CHUNKS) * (TT * KT);
    const int c0    = blockIdx.y * CB;

    const float*   xbase    = x + ((size_t)b * NT) * ND + c0;
    const uint32_t lds_base = (uint32_t)(uintptr_t)(&tile[0]);
    const bool     wave0    = (tid >> 5) == 0;

    // ---- weights for this thread's channel pair, loaded once -------------
    const int cl = tid * 2;
    const int c  = c0 + cl;
    const v2f vzero = {0.0f, 0.0f};

    v2f w[NS];
    const float* Wr = W + (size_t)c * NS;
#pragma unroll
    for (int j = 0; j < NS; ++j) {
        v2f t;
        t.x = Wr[j];
        t.y = Wr[NS + j];
        w[j] = t;
    }

#if USE_TDM
    // prologue: DMA tile 0 into buffer 0
    if (wave0) {
        const int t0   = tbase;
        const int rows = (NT - t0 < ROWS) ? (NT - t0) : ROWS;
        tdm_issue(xbase + (size_t)t0 * ND, lds_base, rows);
    }
#endif

#pragma unroll 1
    for (int k = 0; k < KT; ++k) {
        const int t0   = tbase + k * TT;
        const int rows = (NT - t0 < ROWS) ? (NT - t0) : ROWS;

#if USE_TDM
        if (wave0) {
            if (k + 1 < KT) {
                // prefetch tile k+1 into the other buffer, then wait until
                // only that DMA is outstanding (=> tile k's DMA is done)
                const int t1   = t0 + TT;
                const int rs1  = (NT - t1 < ROWS) ? (NT - t1) : ROWS;
                tdm_issue(xbase + (size_t)t1 * ND,
                          lds_base + (uint32_t)(((k + 1) & 1) * TILE_ELEMS * 4),
                          rs1);
                __builtin_amdgcn_s_wait_tensorcnt(1);
            } else {
                __builtin_amdgcn_s_wait_tensorcnt(0);
            }
        }
        __syncthreads();                 // buf[k&1] is ready for all waves
#else
        // ---- fallback: per-lane async b128 copies of tile k --------------
        {
            const float* gb = xbase + (size_t)t0 * ND;
            const uint32_t lb = lds_base + (uint32_t)((k & 1) * TILE_ELEMS * 4);
            for (int kk = tid; kk < ROWS * CHUNKS_PER_ROW; kk += THREADS) {
                const int row = kk >> 6;          // CHUNKS_PER_ROW == 64
                const int col = kk & 63;
                if (row < rows) {
                    uint32_t ldsoff = lb + (uint32_t)kk * 16u;
                    uint64_t gaddr  = (uint64_t)(uintptr_t)(gb + (size_t)row * ND + col * 4);
                    asm volatile("global_load_async_to_lds_b128 %0, %1, off"
                                 :: "v"(ldsoff), "v"(gaddr) : "memory");
                }
            }
            asm volatile("s_wait_asynccnt 0" ::: "memory");
        }
        __syncthreads();
#endif

        // ---- compute tile k: 2 channels x TT time steps ------------------
        const float* buf = &tile[(k & 1) * TILE_ELEMS];

        v2f acc[TT];
#pragma unroll
        for (int a = 0; a < TT; ++a) acc[a] = vzero;

#pragma unroll
        for (int i = 0; i < ROWS; ++i) {
            v2f xv = *(const v2f*)(&buf[i * CB + cl]);
            if (i >= rows) xv = vzero;   // zero padding at sequence end
#pragma unroll
            for (int a = 0; a < TT; ++a) {
                if (a <= i && (i - a) < NS) {
                    acc[a] = __builtin_elementwise_fma(xv, w[i - a], acc[a]);
                }
            }
        }

        float* obase = out + ((size_t)b * NT + t0) * ND + c;
#pragma unroll
        for (int a = 0; a < TT; ++a) {
            *(v2f*)(obase + (size_t)a * ND) = acc[a];
        }

        __syncthreads();   // all reads of buf[k&1] done before it is reused
    }
}

extern "C" void kernel_launch(void* const* d_in, const int* in_sizes, int n_in,
                              void* d_out, int out_size, void* d_ws, size_t ws_size,
                              hipStream_t stream) {
    const float* x = (const float*)d_in[0];   // (B, T, D) fp32
    const float* W = (const float*)d_in[1];   // (D, S)   fp32
    float* out = (float*)d_out;               // (B, T, D) fp32

    dim3 grid(NB * (NT / (TT * KT)), ND / CB);   // (256, 4)
    dim3 block(THREADS);                         // 128 threads = 4 waves
    rowconv_tdm_pipe_kernel<<<grid, block, 0, stream>>>(x, W, out);
}